// ViT_37074157699251
// MI455X (gfx1250) — compile-verified
//
#include <hip/hip_runtime.h>
#include <hip/hip_bf16.h>

// ViT patch extraction: [256,384,384,3] f32 -> [256,576,768] f32.
// Pure data movement (0 FLOPs): 906 MB total traffic, HBM-roofline ~39us @ 23.3 TB/s.
//
// Key structural trick: the permutation is block-diagonal per image and the
// per-image footprint is identical on both sides (110,592 float4). Choosing the
// per-thread item stride as a multiple of the image size (n4/ITEMS = 32 images)
// makes src and dst advance by the SAME constant between items:
//     src_k = src_0 + k*stride,  dst_k = dst_0 + k*stride
// -> one address decomposition per thread, then ITEMS back-to-back B128
//    NT loads followed by staggered-wait NT stores (max MLP per wave).

typedef float v4f __attribute__((ext_vector_type(4)));

#define HW          384u   // H == W
#define PATCH       16u
#define NPR         24u    // patches per row/col
#define F4_PER_SEG  12u    // 48 floats (16 cols * 3 ch) per patch-row segment
#define F4_PER_ROW  288u   // 384*3/4 float4 per image row
#define ITEMS       8u

__global__ __launch_bounds__(256) void vit_patchify_nt(const v4f* __restrict__ in,
                                                       v4f* __restrict__ out,
                                                       unsigned n4) {
    const unsigned stride = n4 / ITEMS;        // = 3,538,944 f4 = 32 whole images
    const unsigned t = blockIdx.x * 256u + threadIdx.x;
    if (t >= stride) return;                   // exact grid: never taken

    // Decompose ONCE: t = seg*12 + r ; seg = ((b*24 + p)*24 + q)*16 + i
    unsigned seg = t / F4_PER_SEG;
    unsigned r   = t - seg * F4_PER_SEG;
    unsigned i   = seg & (PATCH - 1u);         // row within patch
    unsigned s2  = seg >> 4;                   // (b*24 + p)*24 + q
    unsigned q   = s2 % NPR;                   // patch column
    unsigned s3  = s2 / NPR;                   // b*24 + p
    unsigned p   = s3 % NPR;                   // patch row
    unsigned b   = s3 / NPR;                   // image
    unsigned row = b * HW + p * PATCH + i;     // global image row
    unsigned src = row * F4_PER_ROW + q * F4_PER_SEG + r;

    // Single basic block: ITEMS loads issue back-to-back, waits stagger down.
    v4f v[ITEMS];
#pragma unroll
    for (unsigned k = 0; k < ITEMS; ++k)
        v[k] = __builtin_nontemporal_load(in + src + k * stride);
#pragma unroll
    for (unsigned k = 0; k < ITEMS; ++k)
        __builtin_nontemporal_store(v[k], out + t + k * stride);
}

extern "C" void kernel_launch(void* const* d_in, const int* in_sizes, int n_in,
                              void* d_out, int out_size, void* d_ws, size_t ws_size,
                              hipStream_t stream) {
    const v4f* in  = (const v4f*)d_in[0];
    v4f*       out = (v4f*)d_out;

    // out_size = 256*576*768 = 113,246,208 floats -> 28,311,552 float4.
    unsigned n4      = (unsigned)(out_size / 4);
    unsigned threads = 256u;
    unsigned nt      = n4 / ITEMS;                       // threads needed (exact)
    unsigned blocks  = (nt + threads - 1u) / threads;    // = 13,824 exactly

    vit_patchify_nt<<<blocks, threads, 0, stream>>>(in, out, n4);
}